// Simpler_GCN_4492535792518
// MI455X (gfx1250) — compile-verified
//
#include <hip/hip_runtime.h>
#include <hip/hip_bf16.h>

#define GAT_SLOPE 0.2f
#define ACT_SLOPE 0.1f

typedef __attribute__((ext_vector_type(2))) float v2f;
typedef __attribute__((ext_vector_type(4))) float v4f;
typedef __attribute__((ext_vector_type(8))) float v8f;

#define WAVES_PER_BLOCK 8
#define TILE_F   400     // 16 rows * 25 feats
#define TILE_PAD 404     // + 4 zeroed slack floats (row-15 K-padding reads)

__device__ __forceinline__ float lrelu(float v, float slope) {
    return v >= 0.f ? v : slope * v;
}

// Correct float atomic max for mixed-sign values with -inf init.
__device__ __forceinline__ void atomicMaxF(float* addr, float val) {
    if (val >= 0.f)
        atomicMax((int*)addr, __float_as_int(val));
    else
        atomicMin((unsigned int*)addr, __float_as_uint(val));
}

// ---------------------------------------------------------------------------
// Pack 6 weight matrices (each IN=25 x OUT=5) into zero-padded Bpad[28][32].
// Columns: [0:5)=Wl_p [5:10)=Wr_p [10:15)=Wl_s [15:20)=Wr_s [20:25)=Wl_v [25:30)=Wr_v
__global__ void pack_weights(const float* __restrict__ Wlp, const float* __restrict__ Wrp,
                             const float* __restrict__ Wls, const float* __restrict__ Wrs,
                             const float* __restrict__ Wlv, const float* __restrict__ Wrv,
                             float* __restrict__ Bpad) {
    int t = blockIdx.x * blockDim.x + threadIdx.x;
    if (t >= 28 * 32) return;
    int k = t >> 5, n = t & 31;
    float v = 0.f;
    if (k < 25 && n < 30) {
        int rel = n / 10, which = (n % 10) / 5, feat = n % 5;
        const float* W = (rel == 0) ? (which ? Wrp : Wlp)
                       : (rel == 1) ? (which ? Wrs : Wls)
                                    : (which ? Wrv : Wlv);
        v = W[k * 5 + feat];
    }
    Bpad[t] = v;
}

// ---------------------------------------------------------------------------
// WMMA feature transform: per wave, one 16-node tile staged through LDS.
// A = x[tile, 0:28] (K zero-padded via Bpad's zero rows 25..27),
// B = Bpad (28x32), D = 16x32 outputs scattered into six packed [N][5]
// arrays (xl_p, xr_p, xl_s, xr_s, xl_v, xr_v).
__global__ void transform_wmma(const float* __restrict__ x,
                               const float* __restrict__ Bpad,
                               float* __restrict__ xfeat, int nTiles, int N) {
    __shared__ float tileX[WAVES_PER_BLOCK * TILE_PAD];
    int wslot = threadIdx.x >> 5;
    int wave  = blockIdx.x * WAVES_PER_BLOCK + wslot;
    if (wave >= nTiles) return;                 // wave-uniform: EXEC stays full
    int lane = threadIdx.x & 31;
    float* lds = tileX + wslot * TILE_PAD;

    // Zero the slack so row-15's k>=25 reads are finite (stale LDS may be NaN).
    if (lane < TILE_PAD - TILE_F) lds[TILE_F + lane] = 0.f;

    // Cooperative coalesced load: 400 floats = 100 float4s (16B aligned).
    size_t tbase = (size_t)wave * TILE_F;
    size_t total = (size_t)N * 25;
#pragma unroll
    for (int i = 0; i < 4; ++i) {
        int idx = lane + i * 32;                // float4 index in tile, 0..99
        if (idx < 100) {
            size_t g = tbase + (size_t)idx * 4;
            if (g + 4 <= total) {
                v4f v = __builtin_nontemporal_load((const v4f*)(x + g));
                lds[idx * 4 + 0] = v.x; lds[idx * 4 + 1] = v.y;
                lds[idx * 4 + 2] = v.z; lds[idx * 4 + 3] = v.w;
            } else {                            // tail tile: zero-fill
                for (int j = 0; j < 4; ++j)
                    lds[idx * 4 + j] = (g + j < total) ? x[g + j] : 0.f;
            }
        }
    }
    // Same-wave LDS RAW only -> dscnt wait handled by compiler; no barrier.

    int mrow  = lane & 15;
    int khalf = lane >> 4;                      // 0 -> K{+0,+1}, 1 -> K{+2,+3}
    const float* arow = lds + mrow * 25;

    v8f c0 = {0.f,0.f,0.f,0.f,0.f,0.f,0.f,0.f};
    v8f c1 = {0.f,0.f,0.f,0.f,0.f,0.f,0.f,0.f};

#pragma unroll
    for (int kk = 0; kk < 7; ++kk) {            // K = 28 in steps of 4
        int k0 = kk * 4 + khalf * 2;
        v2f a;                                  // k>=25 values are finite junk
        a.x = arow[k0];                         // and hit zero B rows -> no-op
        a.y = arow[k0 + 1];
        v2f b0, b1;
        b0.x = Bpad[k0 * 32 + mrow];       b0.y = Bpad[(k0 + 1) * 32 + mrow];
        b1.x = Bpad[k0 * 32 + 16 + mrow];  b1.y = Bpad[(k0 + 1) * 32 + 16 + mrow];
        c0 = __builtin_amdgcn_wmma_f32_16x16x4_f32(false, a, false, b0, (short)0, c0, false, false);
        c1 = __builtin_amdgcn_wmma_f32_16x16x4_f32(false, a, false, b1, (short)0, c1, false, false);
    }

    // D layout: lane L holds column N=(L&15)+tilebase, VGPR r -> row M=r (+8 for L>=16)
    int col0 = mrow;            // 0..15, always a valid output column
    int col1 = 16 + mrow;       // 16..31, valid only if < 30
    int rel0 = col0 / 10, wh0 = (col0 % 10) / 5, f0 = col0 % 5;
    size_t off0 = ((size_t)(rel0 * 2 + wh0)) * (size_t)N * 5 + f0;
    size_t off1 = 0; bool c1ok = (col1 < 30);
    if (c1ok) {
        int rel1 = col1 / 10, wh1 = (col1 % 10) / 5, f1 = col1 % 5;
        off1 = ((size_t)(rel1 * 2 + wh1)) * (size_t)N * 5 + f1;
    }
#pragma unroll
    for (int r = 0; r < 8; ++r) {
        int nodeo = wave * 16 + khalf * 8 + r;
        if (nodeo < N) {
            xfeat[off0 + (size_t)nodeo * 5] = c0[r];
            if (c1ok) xfeat[off1 + (size_t)nodeo * 5] = c1[r];
        }
    }
}

// ---------------------------------------------------------------------------
__global__ void init_m(float* __restrict__ m, int n) {
    int i = blockIdx.x * blockDim.x + threadIdx.x;
    if (i < n) m[i] = -INFINITY;
}

// Edge pass A: per-dst max of attention logits. Index streams are one-shot ->
// non-temporal so the hot xl/xr gather arrays stay resident in the 192MB L2.
__global__ void edge_max(const int* __restrict__ ei, long E,
                         const float* __restrict__ xl, const float* __restrict__ xr,
                         const float* __restrict__ att, float* __restrict__ m) {
    long e = (long)blockIdx.x * blockDim.x + threadIdx.x;
    if (e >= E) return;
    int s = __builtin_nontemporal_load(ei + e);
    int d = __builtin_nontemporal_load(ei + E + e);
    const float* pl = xl + (size_t)s * 5;
    const float* pr = xr + (size_t)d * 5;
    float acc = 0.f;
#pragma unroll
    for (int k = 0; k < 5; ++k)
        acc += lrelu(pl[k] + pr[k], GAT_SLOPE) * att[k];
    atomicMaxF(m + d, acc);
}

// Self-loop: fold e_self into m, seed denom/acc with the self contribution.
__global__ void self_init(const float* __restrict__ xl, const float* __restrict__ xr,
                          const float* __restrict__ att, float* __restrict__ m,
                          float* __restrict__ denom, float* __restrict__ acc5, int n) {
    int i = blockIdx.x * blockDim.x + threadIdx.x;
    if (i >= n) return;
    float xlv[5];
    float es = 0.f;
#pragma unroll
    for (int k = 0; k < 5; ++k) {
        float l = xl[(size_t)i * 5 + k];
        xlv[k] = l;
        es += lrelu(l + xr[(size_t)i * 5 + k], GAT_SLOPE) * att[k];
    }
    float mm = fmaxf(m[i], es);
    m[i] = mm;
    float w = __expf(es - mm);
    denom[i] = w;
#pragma unroll
    for (int k = 0; k < 5; ++k) acc5[(size_t)i * 5 + k] = w * xlv[k];
}

// Edge pass B: accumulate exp(e-m) and exp(e-m)*xl[src] into dst.
__global__ void edge_acc(const int* __restrict__ ei, long E,
                         const float* __restrict__ xl, const float* __restrict__ xr,
                         const float* __restrict__ att, const float* __restrict__ m,
                         float* __restrict__ denom, float* __restrict__ acc5) {
    long e = (long)blockIdx.x * blockDim.x + threadIdx.x;
    if (e >= E) return;
    int s = __builtin_nontemporal_load(ei + e);
    int d = __builtin_nontemporal_load(ei + E + e);
    const float* pl = xl + (size_t)s * 5;
    const float* pr = xr + (size_t)d * 5;
    float xlv[5];
    float elog = 0.f;
#pragma unroll
    for (int k = 0; k < 5; ++k) {
        float l = pl[k];
        xlv[k] = l;
        elog += lrelu(l + pr[k], GAT_SLOPE) * att[k];
    }
    float ex = __expf(elog - m[d]);
    atomicAdd(denom + d, ex);
#pragma unroll
    for (int k = 0; k < 5; ++k)
        atomicAdd(acc5 + (size_t)d * 5 + k, ex * xlv[k]);
}

// out = acc/denom + bias, LeakyReLU(0.1), write concat slice of xcat[N][15].
// NT stores: xcat is read once later; don't evict xl/xr from L2 now.
__global__ void norm_relu(const float* __restrict__ denom, const float* __restrict__ acc5,
                          const float* __restrict__ bias, float* __restrict__ xcat,
                          int rel, int n) {
    int i = blockIdx.x * blockDim.x + threadIdx.x;
    if (i >= n) return;
    float inv = 1.f / denom[i];
#pragma unroll
    for (int k = 0; k < 5; ++k) {
        float v = acc5[(size_t)i * 5 + k] * inv + bias[k];
        __builtin_nontemporal_store(lrelu(v, ACT_SLOPE),
                                    xcat + (size_t)i * 15 + rel * 5 + k);
    }
}

// Projection + classifier MLP: 15 -> 10 (lrelu) -> 5 -> 5 (lrelu) -> 2.
__global__ void mlp_head(const float* __restrict__ xcat,
                         const float* __restrict__ Wp1, const float* __restrict__ bp1,
                         const float* __restrict__ Wp2, const float* __restrict__ bp2,
                         const float* __restrict__ Wc1, const float* __restrict__ bc1,
                         const float* __restrict__ Wc2, const float* __restrict__ bc2,
                         float* __restrict__ out, int n) {
    int i = blockIdx.x * blockDim.x + threadIdx.x;
    if (i >= n) return;
    float h0[15];
#pragma unroll
    for (int k = 0; k < 15; ++k)
        h0[k] = __builtin_nontemporal_load(xcat + (size_t)i * 15 + k);
    float h1[10];
#pragma unroll
    for (int j = 0; j < 10; ++j) {
        float s = bp1[j];
#pragma unroll
        for (int k = 0; k < 15; ++k) s += h0[k] * Wp1[k * 10 + j];
        h1[j] = lrelu(s, ACT_SLOPE);
    }
    float h2[5];
#pragma unroll
    for (int j = 0; j < 5; ++j) {
        float s = bp2[j];
#pragma unroll
        for (int k = 0; k < 10; ++k) s += h1[k] * Wp2[k * 5 + j];
        h2[j] = s;
    }
    float h3[5];
#pragma unroll
    for (int j = 0; j < 5; ++j) {
        float s = bc1[j];
#pragma unroll
        for (int k = 0; k < 5; ++k) s += h2[k] * Wc1[k * 5 + j];
        h3[j] = lrelu(s, ACT_SLOPE);
    }
#pragma unroll
    for (int j = 0; j < 2; ++j) {
        float s = bc2[j];
#pragma unroll
        for (int k = 0; k < 5; ++k) s += h3[k] * Wc2[k * 2 + j];
        out[(size_t)i * 2 + j] = s;
    }
}

// ---------------------------------------------------------------------------
extern "C" void kernel_launch(void* const* d_in, const int* in_sizes, int n_in,
                              void* d_out, int out_size, void* d_ws, size_t ws_size,
                              hipStream_t stream) {
    const float* x     = (const float*)d_in[0];
    const int*   ei[3] = {(const int*)d_in[1], (const int*)d_in[2], (const int*)d_in[3]};
    const int  N = in_sizes[0] / 25;
    const long E = in_sizes[1] / 2;

    // Workspace layout (floats):
    //   xfeat : 6 * N * 5   (xl_p, xr_p, xl_s, xr_s, xl_v, xr_v; packed -> fits L2)
    //   m     : N
    //   denom : N
    //   acc   : 5 * N
    //   xcat  : 15 * N
    //   Bpad  : 28 * 32
    float* ws    = (float*)d_ws;
    float* xfeat = ws;
    float* m     = xfeat + (size_t)30 * N;
    float* denom = m + N;
    float* acc   = denom + N;
    float* xcat  = acc + (size_t)5 * N;
    float* Bpad  = xcat + (size_t)15 * N;

    pack_weights<<<4, 256, 0, stream>>>(
        (const float*)d_in[4],  (const float*)d_in[5],
        (const float*)d_in[8],  (const float*)d_in[9],
        (const float*)d_in[12], (const float*)d_in[13], Bpad);

    int nTiles  = (N + 15) / 16;
    int tBlocks = (nTiles + WAVES_PER_BLOCK - 1) / WAVES_PER_BLOCK;
    transform_wmma<<<tBlocks, 32 * WAVES_PER_BLOCK, 0, stream>>>(x, Bpad, xfeat, nTiles, N);

    int nBlocks = (N + 255) / 256;
    int eBlocks = (int)((E + 255) / 256);
    for (int r = 0; r < 3; ++r) {
        const float* xl   = xfeat + (size_t)(2 * r)     * 5 * N;
        const float* xr   = xfeat + (size_t)(2 * r + 1) * 5 * N;
        const float* att  = (const float*)d_in[6 + 4 * r];
        const float* bias = (const float*)d_in[7 + 4 * r];
        init_m   <<<nBlocks, 256, 0, stream>>>(m, N);
        edge_max <<<eBlocks, 256, 0, stream>>>(ei[r], E, xl, xr, att, m);
        self_init<<<nBlocks, 256, 0, stream>>>(xl, xr, att, m, denom, acc, N);
        edge_acc <<<eBlocks, 256, 0, stream>>>(ei[r], E, xl, xr, att, m, denom, acc);
        norm_relu<<<nBlocks, 256, 0, stream>>>(denom, acc, bias, xcat, r, N);
    }

    mlp_head<<<nBlocks, 256, 0, stream>>>(
        xcat,
        (const float*)d_in[16], (const float*)d_in[17],
        (const float*)d_in[18], (const float*)d_in[19],
        (const float*)d_in[20], (const float*)d_in[21],
        (const float*)d_in[22], (const float*)d_in[23],
        (float*)d_out, N);
}